// GNN_11742440587666
// MI455X (gfx1250) — compile-verified
//
#include <hip/hip_runtime.h>
#include <math.h>

// ---------------------------------------------------------------------------
// Quantum GNN on MI455X (gfx1250, wave32)
//  - index extraction from one-hot Ri/Ro  (dense matmul -> gather)
//  - 8-qubit edge circuit : 1 wave/edge, 8 amps/lane in VGPRs
//  - 12-qubit node circuit: 1 wave/node, 128 amps/lane in VGPRs
//  - logical qubit bits remapped so the least-used bits live on lanes:
//      only 5 cross-lane RYs per node circuit; all CNOTs are register
//      renames (hh) or cndmask selects (lh) -- no LDS traffic for CNOTs
//  - mi/mo = (Ri*e)@bo, (Ro*e)@bi via V_WMMA_F32_16X16X4_F32
// ---------------------------------------------------------------------------

typedef __attribute__((ext_vector_type(2))) float v2f;
typedef __attribute__((ext_vector_type(8))) float v8f;

#define TWO_PI_F 6.283185307179586f
#define PI_F     3.14159265358979323846f

// ----------------------------- gate helpers --------------------------------
// amplitude index = r*32 + lane ; lane holds 5 remapped bits, r the rest

template <int R, int RB>
__device__ __forceinline__ void ry_hi(float* a, float c, float s) {
#pragma unroll
  for (int r = 0; r < R; ++r) {
    if ((r & (1 << RB)) == 0) {
      const int r1 = r | (1 << RB);
      float a0 = a[r], a1 = a[r1];
      a[r]  = c * a0 - s * a1;
      a[r1] = s * a0 + c * a1;
    }
  }
}

template <int R, int BIT>
__device__ __forceinline__ void ry_lo(float* a, float c, float s, int lane) {
  const float sgn = ((lane >> BIT) & 1) ? s : -s;
#pragma unroll
  for (int r = 0; r < R; ++r) {
    float p = __shfl_xor(a[r], 1 << BIT, 32);
    a[r] = fmaf(sgn, p, c * a[r]);
  }
}

template <int R, int CRB, int TRB>
__device__ __forceinline__ void cx_hh(float* a) {
#pragma unroll
  for (int r = 0; r < R; ++r) {
    if (((r >> CRB) & 1) && !((r >> TRB) & 1)) {
      const int r1 = r | (1 << TRB);
      float t = a[r]; a[r] = a[r1]; a[r1] = t;   // pure register rename
    }
  }
}

template <int R, int CBIT, int TRB>
__device__ __forceinline__ void cx_lh(float* a, int lane) {
  const bool c1 = (lane >> CBIT) & 1;
#pragma unroll
  for (int r = 0; r < R; ++r) {
    if (!((r >> TRB) & 1)) {
      const int r1 = r | (1 << TRB);
      float a0 = a[r], a1 = a[r1];
      a[r]  = c1 ? a1 : a0;                       // v_cndmask, no shuffle
      a[r1] = c1 ? a0 : a1;
    }
  }
}

template <int R, int CBIT, int TBIT>
__device__ __forceinline__ void cx_ll(float* a, int lane) {
  const bool c1 = (lane >> CBIT) & 1;
#pragma unroll
  for (int r = 0; r < R; ++r) {
    float p = __shfl_xor(a[r], 1 << TBIT, 32);
    a[r] = c1 ? p : a[r];
  }
}

__device__ __forceinline__ float wave_sum32(float v) {
#pragma unroll
  for (int off = 16; off >= 1; off >>= 1) v += __shfl_xor(v, off, 32);
  return v;
}

// just-in-time theta trig: each theta feeds exactly one gate
#define RY_HI_T(RB, TH) do { float h_ = 0.5f * (TH); ry_hi<R, RB>(a, cosf(h_), sinf(h_)); } while (0)
#define RY_LO_T(LB, TH) do { float h_ = 0.5f * (TH); ry_lo<R, LB>(a, cosf(h_), sinf(h_), lane); } while (0)

// ------------------------- index extraction --------------------------------
__global__ void k_indices(const float* __restrict__ Ri,
                          const float* __restrict__ Ro,
                          int* __restrict__ recv, int* __restrict__ send,
                          int N, int E) {
  int e = blockIdx.x * blockDim.x + threadIdx.x;
  if (e >= E) return;
  int r = 0, s = 0;
  for (int n = 0; n < N; ++n) {
    if (Ri[(size_t)n * E + e] != 0.0f) r = n;
    if (Ro[(size_t)n * E + e] != 0.0f) s = n;
  }
  recv[e] = r;
  send[e] = s;
}

// -------------------------------- H0 ---------------------------------------
__global__ void k_h0(const float* __restrict__ X, const float* __restrict__ W,
                     const float* __restrict__ b, float* __restrict__ H, int N) {
  int n = blockIdx.x * blockDim.x + threadIdx.x;
  if (n >= N) return;
  float z = X[n * 3 + 0] * W[0] + X[n * 3 + 1] * W[1] + X[n * 3 + 2] * W[2] + b[0];
  float sig = 1.0f / (1.0f + expf(-z));
  H[n * 4 + 0] = sig * TWO_PI_F;
  H[n * 4 + 1] = X[n * 3 + 0];
  H[n * 4 + 2] = X[n * 3 + 1];
  H[n * 4 + 3] = X[n * 3 + 2];
}

// ----------------------- 8-qubit edge circuit ------------------------------
// wire w -> logical bit (7-w); angle for bit b = ang[7-b]
// lane bits: b0->l0, b1->l1, b3->l2, b4->l3, b7->l4 ; reg: b2->r0, b5->r1, b6->r2
// measure wire 5 = b2 -> r0
__global__ void k_edge(const float* __restrict__ H,
                       const int* __restrict__ recv, const int* __restrict__ send,
                       const float* __restrict__ te,
                       float* __restrict__ eout,
                       float* __restrict__ Bo, float* __restrict__ Bi, int E) {
  const int wid  = (blockIdx.x * blockDim.x + threadIdx.x) >> 5;
  const int lane = threadIdx.x & 31;
  if (wid >= E) return;
  const int sn = send[wid], rn = recv[wid];

  if (lane < 4) {
    Bo[wid * 4 + lane] = H[sn * 4 + lane];
    Bi[wid * 4 + lane] = H[rn * 4 + lane];
  }

  float ang[8];
#pragma unroll
  for (int j = 0; j < 4; ++j) { ang[j] = H[sn * 4 + j]; ang[4 + j] = H[rn * 4 + j]; }

  constexpr int laneLog[5] = {0, 1, 3, 4, 7};
  constexpr int regLog[3]  = {2, 5, 6};
  constexpr int R = 8;

  // tensor-product initial state (replaces 8 encoding RY gate applications)
  float plane = 1.0f;
#pragma unroll
  for (int k = 0; k < 5; ++k) {
    float h = 0.5f * ang[7 - laneLog[k]];
    plane *= ((lane >> k) & 1) ? sinf(h) : cosf(h);
  }
  float a[R];
  a[0] = plane;
#pragma unroll
  for (int rb = 0; rb < 3; ++rb) {
    float h = 0.5f * ang[7 - regLog[rb]];
    const float cc = cosf(h), ss = sinf(h);
#pragma unroll
    for (int r = 0; r < (1 << rb); ++r) { a[r + (1 << rb)] = a[r] * ss; a[r] *= cc; }
  }

  RY_LO_T(4, te[0]);            // w0 (b7)
  RY_HI_T(2, te[1]);            // w1 (b6)
  cx_lh<R, 4, 2>(a, lane);      // cx 0->1
  RY_HI_T(1, te[2]);            // w2 (b5)
  RY_LO_T(3, te[3]);            // w3 (b4)
  cx_lh<R, 3, 1>(a, lane);      // cx 3->2
  RY_LO_T(2, te[4]);            // w4 (b3)
  RY_HI_T(0, te[5]);            // w5 (b2)
  cx_lh<R, 2, 0>(a, lane);      // cx 4->5
  RY_LO_T(1, te[6]);            // w6 (b1)
  RY_LO_T(0, te[7]);            // w7 (b0)
  cx_ll<R, 0, 1>(a, lane);      // cx 7->6
  RY_HI_T(2, te[8]);            // w1
  RY_HI_T(1, te[9]);            // w2
  cx_hh<R, 2, 1>(a);            // cx 1->2
  RY_HI_T(0, te[10]);           // w5
  RY_LO_T(1, te[11]);           // w6
  cx_lh<R, 1, 0>(a, lane);      // cx 6->5
  RY_HI_T(1, te[12]);           // w2
  RY_HI_T(0, te[13]);           // w5
  cx_hh<R, 1, 0>(a);            // cx 2->5
  RY_HI_T(0, te[14]);           // w5

  float part = 0.0f;
#pragma unroll
  for (int r = 0; r < R; ++r) {
    const float sg = (r & 1) ? -1.0f : 1.0f;    // measured bit = r0
    part += a[r] * a[r] * sg;
  }
  float total = wave_sum32(part);
  if (lane == 0) eout[wid] = 0.5f * (1.0f - total);
}

// -------------------- mi/mo via WMMA f32 16x16x4 ----------------------------
// mi = (Ri . diag(e)) @ Bo ; mo = (Ro . diag(e)) @ Bi ; one wave per 16 nodes
__global__ void k_mimo(const float* __restrict__ Ri, const float* __restrict__ Ro,
                       const float* __restrict__ ebuf,
                       const float* __restrict__ Bo, const float* __restrict__ Bi,
                       float* __restrict__ mi, float* __restrict__ mo, int E) {
  const int tile = blockIdx.x;
  const int lane = threadIdx.x;           // 32 threads, all active (EXEC all-1)
  const int m    = tile * 16 + (lane & 15);
  const int klo  = (lane >> 4) * 2;       // A/B K-pair held by this half-wave
  const int n4   = lane & 15;             // output column
  const float msk = (n4 < 4) ? 1.0f : 0.0f;
  const int   nc  = n4 & 3;               // clamped column: no divergent loads

  const float* __restrict__ riRow = Ri + (size_t)m * E;
  const float* __restrict__ roRow = Ro + (size_t)m * E;

  v8f acc_i = {};
  v8f acc_o = {};
  for (int k4 = 0; k4 < E; k4 += 4) {
    const int k0 = k4 + klo, k1 = k0 + 1;
    __builtin_prefetch(&riRow[k0 + 256], 0, 1);
    __builtin_prefetch(&roRow[k0 + 256], 0, 1);
    const float e0 = ebuf[k0], e1 = ebuf[k1];
    v2f a_i = {riRow[k0] * e0, riRow[k1] * e1};
    v2f a_o = {roRow[k0] * e0, roRow[k1] * e1};
    v2f b_i = {Bo[k0 * 4 + nc] * msk, Bo[k1 * 4 + nc] * msk};
    v2f b_o = {Bi[k0 * 4 + nc] * msk, Bi[k1 * 4 + nc] * msk};
    acc_i = __builtin_amdgcn_wmma_f32_16x16x4_f32(false, a_i, false, b_i,
                                                  (short)0, acc_i, false, false);
    acc_o = __builtin_amdgcn_wmma_f32_16x16x4_f32(false, a_o, false, b_o,
                                                  (short)0, acc_o, false, false);
  }
#pragma unroll
  for (int i = 0; i < 8; ++i) {
    const int mrow = tile * 16 + ((lane < 16) ? i : i + 8);
    if (n4 < 4) {
      mi[mrow * 4 + n4] = acc_i[i];
      mo[mrow * 4 + n4] = acc_o[i];
    }
  }
}

// ----------------------- 12-qubit node circuit ------------------------------
// wire w -> logical bit (11-w); angle for bit b = M[11-b]
// lane bits: b0->l0, b3->l1, b4->l2, b8->l3, b11->l4
// reg  bits: b1->r0, b2->r1, b5->r2, b6->r3, b7->r4, b9->r5, b10->r6
// measure wire 9 = b2 -> r1 ; writes H[n*4+0] = pi*(1-<Z>)
__global__ void k_node(const float* __restrict__ mi, const float* __restrict__ mo,
                       float* __restrict__ H, const float* __restrict__ tn, int N) {
  const int wid  = (blockIdx.x * blockDim.x + threadIdx.x) >> 5;
  const int lane = threadIdx.x & 31;
  if (wid >= N) return;

  float ang[12];  // M = [mi(w0..3), mo(w4..7), H(w8..11)]
#pragma unroll
  for (int j = 0; j < 4; ++j) {
    ang[j]     = mi[wid * 4 + j];
    ang[4 + j] = mo[wid * 4 + j];
    ang[8 + j] = H[wid * 4 + j];
  }

  constexpr int laneLog[5] = {0, 3, 4, 8, 11};
  constexpr int regLog[7]  = {1, 2, 5, 6, 7, 9, 10};
  constexpr int R = 128;

  // tensor-product initial state (replaces 12 encoding RY gate applications)
  float plane = 1.0f;
#pragma unroll
  for (int k = 0; k < 5; ++k) {
    float h = 0.5f * ang[11 - laneLog[k]];
    plane *= ((lane >> k) & 1) ? sinf(h) : cosf(h);
  }
  float a[R];
  a[0] = plane;
#pragma unroll
  for (int rb = 0; rb < 7; ++rb) {
    float h = 0.5f * ang[11 - regLog[rb]];
    const float cc = cosf(h), ss = sinf(h);
#pragma unroll
    for (int r = 0; r < (1 << rb); ++r) { a[r + (1 << rb)] = a[r] * ss; a[r] *= cc; }
  }

  RY_LO_T(4, tn[0]);            // w0  (b11)
  RY_HI_T(6, tn[1]);            // w1  (b10)
  cx_lh<R, 4, 6>(a, lane);      // cx 0->1
  RY_HI_T(5, tn[2]);            // w2  (b9)
  RY_LO_T(3, tn[3]);            // w3  (b8)
  cx_lh<R, 3, 5>(a, lane);      // cx 3->2
  RY_HI_T(4, tn[4]);            // w4  (b7)
  RY_HI_T(3, tn[5]);            // w5  (b6)
  cx_hh<R, 4, 3>(a);            // cx 4->5
  RY_HI_T(2, tn[6]);            // w6  (b5)
  RY_LO_T(2, tn[7]);            // w7  (b4)
  cx_lh<R, 2, 2>(a, lane);      // cx 7->6
  RY_LO_T(1, tn[8]);            // w8  (b3)
  RY_HI_T(1, tn[9]);            // w9  (b2)
  cx_lh<R, 1, 1>(a, lane);      // cx 8->9
  RY_HI_T(0, tn[10]);           // w10 (b1)
  RY_LO_T(0, tn[11]);           // w11 (b0)
  cx_lh<R, 0, 0>(a, lane);      // cx 11->10
  RY_HI_T(6, tn[12]);           // w1
  RY_HI_T(5, tn[13]);           // w2
  cx_hh<R, 6, 5>(a);            // cx 1->2
  RY_HI_T(3, tn[14]);           // w5
  RY_HI_T(2, tn[15]);           // w6
  cx_hh<R, 2, 3>(a);            // cx 6->5
  RY_HI_T(1, tn[16]);           // w9
  RY_HI_T(0, tn[17]);           // w10
  cx_hh<R, 0, 1>(a);            // cx 10->9
  RY_HI_T(5, tn[18]);           // w2
  RY_HI_T(3, tn[19]);           // w5
  cx_hh<R, 5, 3>(a);            // cx 2->5
  RY_HI_T(3, tn[20]);           // w5
  RY_HI_T(1, tn[21]);           // w9
  cx_hh<R, 3, 1>(a);            // cx 5->9
  RY_HI_T(4, tn[22]);           // w4

  float part = 0.0f;
#pragma unroll
  for (int r = 0; r < R; ++r) {
    const float sg = ((r >> 1) & 1) ? -1.0f : 1.0f;   // measured bit = r1
    part += a[r] * a[r] * sg;
  }
  float total = wave_sum32(part);
  if (lane == 0) H[wid * 4 + 0] = PI_F * (1.0f - total);
}

// ------------------------------- launch -------------------------------------
extern "C" void kernel_launch(void* const* d_in, const int* in_sizes, int n_in,
                              void* d_out, int out_size, void* d_ws, size_t ws_size,
                              hipStream_t stream) {
  (void)n_in; (void)out_size; (void)ws_size;
  const float* X  = (const float*)d_in[0];
  const float* Ri = (const float*)d_in[1];
  const float* Ro = (const float*)d_in[2];
  const float* W  = (const float*)d_in[3];
  const float* b  = (const float*)d_in[4];
  const float* te = (const float*)d_in[5];
  const float* tn = (const float*)d_in[6];

  const int N = in_sizes[0] / 3;            // 2048
  const int E = in_sizes[1] / N;            // 4096

  char* w = (char*)d_ws;
  int*   recv = (int*)w;   w += (size_t)E * sizeof(int);
  int*   send = (int*)w;   w += (size_t)E * sizeof(int);
  float* H    = (float*)w; w += (size_t)N * 4 * sizeof(float);
  float* Bo   = (float*)w; w += (size_t)E * 4 * sizeof(float);
  float* Bi   = (float*)w; w += (size_t)E * 4 * sizeof(float);
  float* ebuf = (float*)w; w += (size_t)E * sizeof(float);
  float* mi   = (float*)w; w += (size_t)N * 4 * sizeof(float);
  float* mo   = (float*)w; w += (size_t)N * 4 * sizeof(float);

  k_indices<<<(E + 255) / 256, 256, 0, stream>>>(Ri, Ro, recv, send, N, E);
  k_h0<<<(N + 255) / 256, 256, 0, stream>>>(X, W, b, H, N);
  k_edge<<<(E * 32 + 255) / 256, 256, 0, stream>>>(H, recv, send, te, ebuf, Bo, Bi, E);

  for (int it = 0; it < 2; ++it) {
    k_mimo<<<N / 16, 32, 0, stream>>>(Ri, Ro, ebuf, Bo, Bi, mi, mo, E);
    k_node<<<(N * 32 + 255) / 256, 256, 0, stream>>>(mi, mo, H, tn, N);
    float* eo = (it == 1) ? (float*)d_out : ebuf;
    k_edge<<<(E * 32 + 255) / 256, 256, 0, stream>>>(H, recv, send, te, eo, Bo, Bi, E);
  }
}